// FusionModule_1632087573187
// MI455X (gfx1250) — compile-verified
//
#include <hip/hip_runtime.h>

// Problem constants (reference: B,C,T,FR = 4,64,256,257; EMB=256, H=4, CHUNK=64)
#define BB   4
#define CC   64
#define TT   256
#define FRR  257
#define EMBN 256
#define HH   4
#define DHN  16
#define CSN  64
#define NCN  4

typedef float v2f __attribute__((ext_vector_type(2)));
typedef float v8f __attribute__((ext_vector_type(8)));

// fp32 WMMA 16x16x4: D = A(16x4) * B(4x16) + C, wave32.
// A frag: lane l -> A[M = l&15][K = 2*(l>>4)+j], j=0,1
// B frag: lane l -> B[K = 2*(l>>4)+j][N = l&15]
// C/D:    vgpr r, lane l -> [M = r + 8*(l>>4)][N = l&15]
__device__ __forceinline__ v8f wmma_f32_4(v2f a, v2f b, v8f c) {
  return __builtin_amdgcn_wmma_f32_16x16x4_f32(false, a, false, b, (short)0, c,
                                               false, false);
}

// workspace layout (float offsets)
#define WS_QK   0        // [B][H][64]  : scale * q . wk  (folded)
#define WS_W2O  1024     // [64][64]    : Fw2 @ wo
#define WS_CB   5120     // [B][64]     : fusion_b + Fw2 @ aux_p
#define WS_BIAS 5376     // [H][64][64] : rel_bias gathered
// total 21760 floats = 87KB

// ---------------------------------------------------------------------------
// Kernel A: tiny precompute (single block)
// ---------------------------------------------------------------------------
__global__ __launch_bounds__(256) void kA(const float* __restrict__ aux,
                                          const float* __restrict__ lin_w,
                                          const float* __restrict__ wq,
                                          const float* __restrict__ wk,
                                          const float* __restrict__ wo,
                                          const float* __restrict__ rel_bias,
                                          const float* __restrict__ fusion_w,
                                          const float* __restrict__ fusion_b,
                                          float* __restrict__ ws) {
  __shared__ float sAux[BB][CC];
  __shared__ float sQ[BB][CC];
  int tid = threadIdx.x;
  int b = tid >> 6, c = tid & 63;

  // aux_p = aux @ lin_w.T   [B,64]
  float acc = 0.f;
  for (int e = 0; e < EMBN; ++e) acc += aux[b * EMBN + e] * lin_w[c * EMBN + e];
  sAux[b][c] = acc;
  __syncthreads();

  // q = aux_p @ wq.T        [B,64]
  acc = 0.f;
  for (int c2 = 0; c2 < CC; ++c2) acc += sAux[b][c2] * wq[c * CC + c2];
  sQ[b][c] = acc;
  __syncthreads();

  // qk[b][h][c] = scale * sum_d q[h*16+d] * wk[h*16+d][c], scale = 1/sqrt(16)
  for (int i = tid; i < BB * HH * CC; i += 256) {
    int b2 = i >> 8, h = (i >> 6) & 3, cc2 = i & 63;
    float a2 = 0.f;
    for (int d = 0; d < DHN; ++d)
      a2 += sQ[b2][h * DHN + d] * wk[(h * DHN + d) * CC + cc2];
    ws[WS_QK + i] = a2 * 0.25f;
  }
  // W2o[o][c'] = sum_c fusion_w[o][64+c] * wo[c][c']
  for (int i = tid; i < CC * CC; i += 256) {
    int o = i >> 6, cp = i & 63;
    float a2 = 0.f;
    for (int c2 = 0; c2 < CC; ++c2)
      a2 += fusion_w[o * (2 * CC) + CC + c2] * wo[c2 * CC + cp];
    ws[WS_W2O + i] = a2;
  }
  // const_b[b][o] = fusion_b[o] + sum_c fusion_w[o][64+c] * aux_p[b][c]
  {
    int b2 = tid >> 6, o = tid & 63;
    float a2 = fusion_b[o];
    for (int c2 = 0; c2 < CC; ++c2)
      a2 += fusion_w[o * (2 * CC) + CC + c2] * sAux[b2][c2];
    ws[WS_CB + tid] = a2;
  }
  // bias[h][qi][ki] = rel_bias[qi-ki+T-1][h]
  for (int i = tid; i < HH * CSN * CSN; i += 256) {
    int h = i >> 12, qi = (i >> 6) & 63, ki = i & 63;
    ws[WS_BIAS + i] = rel_bias[(qi - ki + TT - 1) * HH + h];
  }
}

// ---------------------------------------------------------------------------
// Kernel B: out = Fw1 @ E + const_b  (per (b,t): 64x64 @ 64x257), f32 WMMA
// grid = B*T (1024), block = 128 (4 waves, wave = o-tile)
// All B-frag loads unconditional (f clamped); dead columns masked at store.
// ---------------------------------------------------------------------------
__global__ __launch_bounds__(128) void kB(const float* __restrict__ esti,
                                          const float* __restrict__ fusion_w,
                                          const float* __restrict__ ws,
                                          float* __restrict__ out) {
  int b = blockIdx.x >> 8, t = blockIdx.x & 255;
  int lane = threadIdx.x & 31, w = threadIdx.x >> 5;
  int lm = lane & 15, lh = lane >> 4;
  const int cstride = TT * FRR;

  int row = w * 16 + lm;  // output channel for A rows
  v2f a[16];
#pragma unroll
  for (int s = 0; s < 16; ++s) {
    int c0 = 4 * s + 2 * lh;
    a[s].x = fusion_w[row * 128 + c0];
    a[s].y = fusion_w[row * 128 + c0 + 1];
  }
  float cb[8];
#pragma unroll
  for (int r = 0; r < 8; ++r) cb[r] = ws[WS_CB + b * 64 + w * 16 + r + 8 * lh];

  int ebase = (b * 64 * TT + t) * FRR;  // esti[b][0][t][0]
  for (int ft = 0; ft < 17; ++ft) {
    int f = ft * 16 + lm;
    int fc = f < FRR ? f : (FRR - 1);  // clamped address, unconditional loads
    bool v = f < FRR;
    const float* ecol = esti + ebase + fc;
    v8f acc = {};
#pragma unroll
    for (int s = 0; s < 16; ++s) {
      int c0 = 4 * s + 2 * lh;
      v2f bb;
      bb.x = ecol[c0 * cstride];
      bb.y = ecol[(c0 + 1) * cstride];
      acc = wmma_f32_4(a[s], bb, acc);
    }
    if (v) {
#pragma unroll
      for (int r = 0; r < 8; ++r) {
        int o = w * 16 + r + 8 * lh;
        out[(b * 64 + o) * cstride + t * FRR + f] = acc[r] + cb[r];
      }
    }
  }
}

// ---------------------------------------------------------------------------
// Kernel C: attention + W2o fold, out += W2o @ (attn . V)
// grid = B * NC * NFT (528), block = 256 (8 waves). f-tile = 8.
// ---------------------------------------------------------------------------
#define FTW  8
#define NFT  33
#define SVS  (FTW * DHN + 1)  // 129, padded for LDS bank spread

__global__ __launch_bounds__(256) void kC(const float* __restrict__ esti,
                                          const float* __restrict__ wv,
                                          const float* __restrict__ ws,
                                          float* __restrict__ out) {
  int bid = blockIdx.x;
  int ft = bid % NFT;
  int n = (bid / NFT) & 3;
  int b = bid / (NFT * 4);
  int t0 = n * CSN, f0 = ft * FTW;
  int tid = threadIdx.x;
  int lane = tid & 31, w = tid >> 5;
  int lm = lane & 15, lh = lane >> 4;
  int fl = lane & 7, kq = lane >> 3;  // softmax mapping: 8 f x 4 ki-quarters
  const int cstride = TT * FRR;

  __shared__ float sS[HH][CSN][FTW + 1];  // base scores (q.k)
  __shared__ float sV[CSN][SVS];          // current head V: [ki][fl*16 + d]
  __shared__ float sOh[8][DHN][FTW];      // per-wave O_h staging

  // ---- Phase 0: S[h][ki][fl] = sum_c qk[b][h][c] * E[c][ki][fl]
  // Addresses clamped; garbage (finite) values for the 7 out-of-range f
  // column flow through softmax bounded (exp<=1) and are masked at store.
  {
    int fl0 = tid & 7;
    int ks = tid >> 3;  // 0..31
    int f = f0 + fl0;
    int fc = f < FRR ? f : (FRR - 1);
    for (int rep = 0; rep < 2; ++rep) {
      int ki = ks + 32 * rep;
      float a0 = 0.f, a1 = 0.f, a2 = 0.f, a3 = 0.f;
      const float* ecol = esti + (b * 64 * TT + (t0 + ki)) * FRR + fc;
      for (int c = 0; c < CC; ++c) {
        float e = ecol[c * cstride];
        a0 += ws[WS_QK + (b * HH + 0) * CC + c] * e;
        a1 += ws[WS_QK + (b * HH + 1) * CC + c] * e;
        a2 += ws[WS_QK + (b * HH + 2) * CC + c] * e;
        a3 += ws[WS_QK + (b * HH + 3) * CC + c] * e;
      }
      sS[0][ki][fl0] = a0;
      sS[1][ki][fl0] = a1;
      sS[2][ki][fl0] = a2;
      sS[3][ki][fl0] = a3;
    }
  }
  __syncthreads();

  int flv = f0 + lm;
  int flc = flv < FRR ? flv : (FRR - 1);  // clamped f for B-frag addresses
  bool vstore = (lm < 8) && (flv < FRR);

  for (int h = 0; h < HH; ++h) {
    __syncthreads();  // protect sV overwrite from previous head's readers
    // ---- V_h[d][ki][f] = wv_h @ E  via WMMA (M=d16, K=c64, N=f)
    for (int i = 0; i < 8; ++i) {
      int ki = w + 8 * i;
      v8f acc = {};
      const float* ecol = esti + (b * 64 * TT + (t0 + ki)) * FRR + flc;
#pragma unroll
      for (int s = 0; s < 16; ++s) {
        int c0 = 4 * s + 2 * lh;
        v2f av, bv;
        av.x = wv[(h * DHN + lm) * CC + c0];
        av.y = wv[(h * DHN + lm) * CC + c0 + 1];
        bv.x = ecol[c0 * cstride];
        bv.y = ecol[(c0 + 1) * cstride];
        acc = wmma_f32_4(av, bv, acc);
      }
      if (lm < 8) {  // store predicate required: row is only 129 floats wide
#pragma unroll
        for (int r = 0; r < 8; ++r) sV[ki][lm * DHN + r + 8 * lh] = acc[r];
      }
    }
    __syncthreads();

    // ---- 8 rounds: wave w owns query qi = r8*8 + w
    for (int r8 = 0; r8 < 8; ++r8) {
      int qi = r8 * 8 + w;
      // softmax over ki (64) per f; lane handles 16 ki
      float p[16];
      float m = -3.0e38f;
#pragma unroll
      for (int i = 0; i < 16; ++i) {
        int ki = kq * 16 + i;
        float sc = sS[h][ki][fl] + ws[WS_BIAS + (h * CSN + qi) * CSN + ki];
        p[i] = sc;
        m = fmaxf(m, sc);
      }
      m = fmaxf(m, __shfl_xor(m, 8));
      m = fmaxf(m, __shfl_xor(m, 16));
      float sum = 0.f;
#pragma unroll
      for (int i = 0; i < 16; ++i) {
        p[i] = __expf(p[i] - m);
        sum += p[i];
      }
      sum += __shfl_xor(sum, 8);
      sum += __shfl_xor(sum, 16);
      float rs = 1.0f / sum;

      // O_h[d][f] = sum_ki p * V_h[d][ki][f]  (VALU partial + butterfly)
      float od[16];
#pragma unroll
      for (int d = 0; d < 16; ++d) od[d] = 0.f;
#pragma unroll
      for (int i = 0; i < 16; ++i) {
        int ki = kq * 16 + i;
        const float* vr = &sV[ki][fl * DHN];
        float pi = p[i];
#pragma unroll
        for (int d = 0; d < 16; ++d) od[d] += pi * vr[d];
      }
#pragma unroll
      for (int d = 0; d < 16; ++d) {
        float x = od[d];
        x += __shfl_xor(x, 8);
        x += __shfl_xor(x, 16);
        od[d] = x * rs;
      }
      if (lane < 8) {
#pragma unroll
        for (int d = 0; d < 16; ++d) sOh[w][d][fl] = od[d];
      }

      // fold: out[o][f] += W2o[:, h*16+d] @ O_h[d][f]  (M=64, K=16, N=f)
      // B-frag columns 8..15 are dead: read duplicated columns (lm&7),
      // unconditional ds loads, no exec masking.
      int lmb = lm & 7;
      v8f f0a = {}, f1a = {}, f2a = {}, f3a = {};
#pragma unroll
      for (int s = 0; s < 4; ++s) {
        int k0 = 4 * s + 2 * lh;
        v2f bo;
        bo.x = sOh[w][k0][lmb];
        bo.y = sOh[w][k0 + 1][lmb];
        v2f aw;
        aw.x = ws[WS_W2O + (0 * 16 + lm) * CC + h * DHN + k0];
        aw.y = ws[WS_W2O + (0 * 16 + lm) * CC + h * DHN + k0 + 1];
        f0a = wmma_f32_4(aw, bo, f0a);
        aw.x = ws[WS_W2O + (1 * 16 + lm) * CC + h * DHN + k0];
        aw.y = ws[WS_W2O + (1 * 16 + lm) * CC + h * DHN + k0 + 1];
        f1a = wmma_f32_4(aw, bo, f1a);
        aw.x = ws[WS_W2O + (2 * 16 + lm) * CC + h * DHN + k0];
        aw.y = ws[WS_W2O + (2 * 16 + lm) * CC + h * DHN + k0 + 1];
        f2a = wmma_f32_4(aw, bo, f2a);
        aw.x = ws[WS_W2O + (3 * 16 + lm) * CC + h * DHN + k0];
        aw.y = ws[WS_W2O + (3 * 16 + lm) * CC + h * DHN + k0 + 1];
        f3a = wmma_f32_4(aw, bo, f3a);
      }
      if (vstore) {
        int obase = (b * 64 * TT + (t0 + qi)) * FRR + f0 + lm;  // o = 0
#pragma unroll
        for (int r = 0; r < 8; ++r) {
          int rr = r + 8 * lh;
          out[obase + (0 * 16 + rr) * cstride] += f0a[r];
          out[obase + (1 * 16 + rr) * cstride] += f1a[r];
          out[obase + (2 * 16 + rr) * cstride] += f2a[r];
          out[obase + (3 * 16 + rr) * cstride] += f3a[r];
        }
      }
    }
  }
}

// ---------------------------------------------------------------------------
extern "C" void kernel_launch(void* const* d_in, const int* in_sizes, int n_in,
                              void* d_out, int out_size, void* d_ws,
                              size_t ws_size, hipStream_t stream) {
  (void)in_sizes; (void)n_in; (void)out_size; (void)ws_size;
  const float* esti     = (const float*)d_in[0];
  const float* aux      = (const float*)d_in[1];
  const float* lin_w    = (const float*)d_in[2];
  const float* wq       = (const float*)d_in[3];
  const float* wk       = (const float*)d_in[4];
  const float* wv       = (const float*)d_in[5];
  const float* wo       = (const float*)d_in[6];
  const float* rel_bias = (const float*)d_in[7];
  const float* fusion_w = (const float*)d_in[8];
  const float* fusion_b = (const float*)d_in[9];
  float* out = (float*)d_out;
  float* ws  = (float*)d_ws;

  kA<<<1, 256, 0, stream>>>(aux, lin_w, wq, wk, wo, rel_bias, fusion_w,
                            fusion_b, ws);
  kB<<<BB * TT, 128, 0, stream>>>(esti, fusion_w, ws, out);
  kC<<<BB * NCN * NFT, 256, 0, stream>>>(esti, wv, ws, out);
}